// EGCN_H_pr_25220047962224
// MI455X (gfx1250) — compile-verified
//
#include <hip/hip_runtime.h>
#include <math.h>

typedef float v2f __attribute__((ext_vector_type(2)));
typedef float v8f __attribute__((ext_vector_type(8)));

#define FDIM 64
#define HDIM 256
#define DOCD 768
#define NOUT 16
#define CHUNK 1024
#define NEG_INF (-__builtin_inff())

// ---------- pool_w / ||pool_w|| ----------
__global__ void prep_pw_kernel(const float* __restrict__ pool_w, float* __restrict__ pw) {
  __shared__ float red[64];
  int t = threadIdx.x;               // 64 threads
  float v = pool_w[t];
  red[t] = v * v;
  __syncthreads();
  for (int s = 32; s > 0; s >>= 1) { if (t < s) red[t] += red[t + s]; __syncthreads(); }
  pw[t] = v * rsqrtf(red[0]);
}

// ---------- score[n] = <x[n], pw> ----------
__global__ void score_kernel(const float* __restrict__ x, const float* __restrict__ pw,
                             float* __restrict__ score, int N) {
  __shared__ float sp[FDIM];
  if (threadIdx.x < FDIM) sp[threadIdx.x] = pw[threadIdx.x];
  __syncthreads();
  int n = blockIdx.x * blockDim.x + threadIdx.x;
  if (n >= N) return;
  const float4* xr = (const float4*)(x + (size_t)n * FDIM);
  const float4* pr = (const float4*)sp;
  float acc = 0.f;
#pragma unroll
  for (int i = 0; i < FDIM / 4; ++i) {
    float4 a = xr[i]; float4 b = pr[i];
    acc += a.x * b.x + a.y * b.y + a.z * b.z + a.w * b.w;
  }
  score[n] = acc;
}

// ---------- deg[col] += 1 over E edges (self-loop folded in later) ----------
__global__ void deg_kernel(const int* __restrict__ col, float* __restrict__ deg, int E) {
  int e = blockIdx.x * blockDim.x + threadIdx.x;
  if (e < E) atomicAdd(&deg[col[e]], 1.0f);
}

__global__ void dis_kernel(const float* __restrict__ deg, float* __restrict__ dis, int N) {
  int n = blockIdx.x * blockDim.x + threadIdx.x;
  if (n < N) dis[n] = rsqrtf(deg[n] + 1.0f);   // +1 = self loop; deg>0 always
}

// ---------- t[row] += dis[col] ----------
__global__ void t_kernel(const int* __restrict__ row, const int* __restrict__ col,
                         const float* __restrict__ dis, float* __restrict__ tb, int E) {
  int e = blockIdx.x * blockDim.x + threadIdx.x;
  if (e < E) atomicAdd(&tb[row[e]], dis[col[e]]);
}

// ---------- svec[f] = sum_n (dis*t + dis^2)[n] * x[n,f] ----------
__global__ void wsum_kernel(const float* __restrict__ x, const float* __restrict__ dis,
                            const float* __restrict__ tb, float* __restrict__ svec, int N) {
  __shared__ float red[256];
  int tid = threadIdx.x;
  int f = tid & 63;
  int rg = tid >> 6;                 // 4 rows per block-iteration
  float acc = 0.f;
  int stride = gridDim.x * 4;
  for (int n = blockIdx.x * 4 + rg; n < N; n += stride) {
    float d = dis[n];
    float c = d * tb[n] + d * d;
    acc += c * x[(size_t)n * FDIM + f];
  }
  red[tid] = acc;
  __syncthreads();
  if (rg == 0) {
    float s = red[f] + red[f + 64] + red[f + 128] + red[f + 192];
    atomicAdd(&svec[f], s);
  }
}

// ---------- top-64: stage 1 (per-block local top-64 of a 1024 chunk) ----------
__global__ void topk_stage1(const float* __restrict__ score, int N,
                            float* __restrict__ cval, int* __restrict__ cidx) {
  __shared__ float sval[CHUNK];
  __shared__ int   sgid[CHUNK];
  __shared__ float rv[256]; __shared__ int rg[256]; __shared__ int rp[256];
  int tid = threadIdx.x;
  int start = blockIdx.x * CHUNK;
  for (int i = tid; i < CHUNK; i += 256) {
    int gi = start + i;
    sval[i] = (gi < N) ? score[gi] : NEG_INF;
    sgid[i] = gi;
  }
  __syncthreads();
  for (int it = 0; it < 64; ++it) {
    float bv = NEG_INF; int bg = 0x7fffffff; int bp = 0;
    for (int i = tid; i < CHUNK; i += 256) {
      float v = sval[i]; int g = sgid[i];
      if (v > bv || (v == bv && g < bg)) { bv = v; bg = g; bp = i; }
    }
    rv[tid] = bv; rg[tid] = bg; rp[tid] = bp;
    __syncthreads();
    for (int s = 128; s > 0; s >>= 1) {
      if (tid < s) {
        float ov = rv[tid + s]; int og = rg[tid + s];
        if (ov > rv[tid] || (ov == rv[tid] && og < rg[tid])) {
          rv[tid] = ov; rg[tid] = og; rp[tid] = rp[tid + s];
        }
      }
      __syncthreads();
    }
    if (tid == 0) {
      cval[blockIdx.x * 64 + it] = rv[0];
      cidx[blockIdx.x * 64 + it] = rg[0];
      sval[rp[0]] = NEG_INF;
    }
    __syncthreads();
  }
}

// ---------- top-64: stage 2 (merge candidates) + build x_tilde ----------
__global__ void topk_stage2(const float* __restrict__ x, float* __restrict__ cval,
                            const int* __restrict__ cidx, int nc,
                            float* __restrict__ x_tilde) {
  __shared__ float rv[256]; __shared__ int rg[256]; __shared__ int rp[256];
  __shared__ float topv[64]; __shared__ int topi[64];
  int tid = threadIdx.x;
  for (int it = 0; it < 64; ++it) {
    float bv = NEG_INF; int bg = 0x7fffffff; int bp = 0;
    for (int i = tid; i < nc; i += 256) {
      float v = cval[i]; int g = cidx[i];
      if (v > bv || (v == bv && g < bg)) { bv = v; bg = g; bp = i; }
    }
    rv[tid] = bv; rg[tid] = bg; rp[tid] = bp;
    __syncthreads();
    for (int s = 128; s > 0; s >>= 1) {
      if (tid < s) {
        float ov = rv[tid + s]; int og = rg[tid + s];
        if (ov > rv[tid] || (ov == rv[tid] && og < rg[tid])) {
          rv[tid] = ov; rg[tid] = og; rp[tid] = rp[tid + s];
        }
      }
      __syncthreads();
    }
    if (tid == 0) {
      topv[it] = rv[0]; topi[it] = rg[0];
      cval[rp[0]] = NEG_INF;           // consume (rewritten next call by stage1)
    }
    __threadfence();
    __syncthreads();
  }
  for (int i = tid; i < 64 * FDIM; i += 256) {
    int r = i >> 6, f = i & 63;
    x_tilde[i] = x[(size_t)topi[r] * FDIM + f] * tanhf(topv[r]);
  }
}

// ---------- GRU gate GEMMs via V_WMMA_F32_16X16X4_F32 ----------
// gx = x_tilde @ w_ih^T + b_ih   [64,192]
// gh = h0      @ w_hh^T + b_hh   [64,192]
__global__ __launch_bounds__(256) void gru_wmma_kernel(
    const float* __restrict__ x_tilde, const float* __restrict__ h0,
    const float* __restrict__ w_ih, const float* __restrict__ w_hh,
    const float* __restrict__ b_ih, const float* __restrict__ b_hh,
    float* __restrict__ gx, float* __restrict__ gh) {
  int tid = threadIdx.x;
  int wave = tid >> 5;
  int lane = tid & 31;
  int m = lane & 15;      // row-in-tile (A) / col-in-tile (B,D)
  int kh = lane >> 4;     // lane-half selects K pair / row-half of D
  // 96 tiles total (2 gemms * 4 Mtiles * 12 Ntiles), 12 per wave, uniform EXEC
  for (int t = wave; t < 96; t += 8) {
    int which = t / 48;
    int tt = t % 48;
    int mi = tt & 3;
    int nj = tt >> 2;
    const float* A   = which ? h0   : x_tilde;  // [64 x 64] row-major
    const float* B   = which ? w_hh : w_ih;     // [192 x 64] row-major (used transposed)
    const float* bia = which ? b_hh : b_ih;
    float* out       = which ? gh   : gx;       // [64 x 192]
    v8f acc = {};
#pragma unroll
    for (int kk = 0; kk < 16; ++kk) {
      int k0 = kk * 4 + 2 * kh;
      v2f a, b;
      // A 16x4 fragment: VGPR0 holds K={0,2}, VGPR1 holds K={1,3} (ISA 7.12.2)
      a[0] = A[(mi * 16 + m) * 64 + k0];
      a[1] = A[(mi * 16 + m) * 64 + k0 + 1];
      // B 4x16 fragment (B[k][n] = w[n][k]), rows split across lane halves
      b[0] = B[(nj * 16 + m) * 64 + k0];
      b[1] = B[(nj * 16 + m) * 64 + k0 + 1];
      acc = __builtin_amdgcn_wmma_f32_16x16x4_f32(false, a, false, b,
                                                  (short)0, acc, false, false);
    }
#pragma unroll
    for (int r = 0; r < 8; ++r) {
      int rowi = mi * 16 + r + 8 * kh;
      int colj = nj * 16 + m;
      out[rowi * 192 + colj] = acc[r] + bia[colj];
    }
  }
}

// ---------- gate math: evolved weight W ----------
__global__ void evolve_kernel(const float* __restrict__ gx, const float* __restrict__ gh,
                              const float* __restrict__ h0, float* __restrict__ W) {
  int idx = blockIdx.x * blockDim.x + threadIdx.x;
  if (idx >= 64 * 64) return;
  int i = idx >> 6, j = idx & 63;
  float r  = 1.f / (1.f + expf(-(gx[i * 192 + j]       + gh[i * 192 + j])));
  float z  = 1.f / (1.f + expf(-(gx[i * 192 + 64 + j]  + gh[i * 192 + 64 + j])));
  float nc = tanhf(gx[i * 192 + 128 + j] + r * gh[i * 192 + 128 + j]);
  W[idx] = (1.f - z) * nc + z * h0[idx];
}

// ---------- tail: pooled -> g, d, layernorm, fusion, heads ----------
__global__ void tail_kernel(const float* __restrict__ svec, const float* __restrict__ W, int N,
                            const float* __restrict__ gnn_w, const float* __restrict__ gnn_b,
                            const float* __restrict__ doc,   const float* __restrict__ doc_w,
                            const float* __restrict__ doc_b,
                            const float* __restrict__ lng,   const float* __restrict__ lnb,
                            const float* __restrict__ fus_w, const float* __restrict__ fus_b,
                            const float* __restrict__ task_w,const float* __restrict__ task_b,
                            const float* __restrict__ time_w,const float* __restrict__ time_b,
                            float* __restrict__ out) {
  __shared__ float pooled[FDIM];
  __shared__ float lnv[2 * HDIM];
  __shared__ float h1[HDIM];
  __shared__ float red[256];
  int t = threadIdx.x;                 // 256 threads
  if (t < FDIM) {
    float s = 0.f;
    for (int k = 0; k < FDIM; ++k) s += svec[k] * W[k * FDIM + t];
    pooled[t] = s / (float)N;          // pooled = (svec @ W) / N
  }
  __syncthreads();
  float gl = gnn_b[t];
  for (int k = 0; k < FDIM; ++k) gl += pooled[k] * gnn_w[t * FDIM + k];
  float dh = doc_b[t];
  for (int k = 0; k < DOCD; ++k) dh += doc[k] * doc_w[t * DOCD + k];
  dh = fmaxf(dh, 0.f);
  // layernorm over fused = [gl(0..255), dh(256..511)]
  red[t] = gl + dh;
  __syncthreads();
  for (int s = 128; s > 0; s >>= 1) { if (t < s) red[t] += red[t + s]; __syncthreads(); }
  float mu = red[0] / 512.f;
  __syncthreads();
  red[t] = (gl - mu) * (gl - mu) + (dh - mu) * (dh - mu);
  __syncthreads();
  for (int s = 128; s > 0; s >>= 1) { if (t < s) red[t] += red[t + s]; __syncthreads(); }
  float rs = rsqrtf(red[0] / 512.f + 1e-5f);
  lnv[t]        = (gl - mu) * rs * lng[t]        + lnb[t];
  lnv[HDIM + t] = (dh - mu) * rs * lng[HDIM + t] + lnb[HDIM + t];
  __syncthreads();
  float h = fus_b[t];
  for (int k = 0; k < 2 * HDIM; ++k) h += lnv[k] * fus_w[t * 2 * HDIM + k];
  h1[t] = fmaxf(h, 0.f);
  __syncthreads();
  if (t < NOUT) {
    float s = task_b[t];
    for (int k = 0; k < HDIM; ++k) s += h1[k] * task_w[t * HDIM + k];
    out[t] = s;
  } else if (t == NOUT) {
    float s = time_b[0];
    for (int k = 0; k < HDIM; ++k) s += h1[k] * time_w[k];
    out[NOUT] = s;
  }
}

extern "C" void kernel_launch(void* const* d_in, const int* in_sizes, int n_in,
                              void* d_out, int out_size, void* d_ws, size_t ws_size,
                              hipStream_t stream) {
  const float* x      = (const float*)d_in[0];
  const int*   ei     = (const int*)  d_in[1];
  const float* doc    = (const float*)d_in[2];
  const float* pool_w = (const float*)d_in[3];
  const float* init_w = (const float*)d_in[4];
  const float* w_ih   = (const float*)d_in[5];
  const float* w_hh   = (const float*)d_in[6];
  const float* b_ih   = (const float*)d_in[7];
  const float* b_hh   = (const float*)d_in[8];
  const float* gnn_w  = (const float*)d_in[9];
  const float* gnn_b  = (const float*)d_in[10];
  const float* doc_w  = (const float*)d_in[11];
  const float* doc_b  = (const float*)d_in[12];
  const float* ln_g   = (const float*)d_in[13];
  const float* ln_b   = (const float*)d_in[14];
  const float* fus_w  = (const float*)d_in[15];
  const float* fus_b  = (const float*)d_in[16];
  const float* task_w = (const float*)d_in[17];
  const float* task_b = (const float*)d_in[18];
  const float* time_w = (const float*)d_in[19];
  const float* time_b = (const float*)d_in[20];
  (void)n_in; (void)ws_size; (void)out_size;

  int N = in_sizes[0] / FDIM;
  int E = in_sizes[1] / 2;
  const int* row = ei;
  const int* col = ei + E;

  float* ws      = (float*)d_ws;
  float* score   = ws;  ws += N;
  float* deg     = ws;  ws += N;
  float* dis     = ws;  ws += N;
  float* tbuf    = ws;  ws += N;
  float* svec    = ws;  ws += 64;
  float* pw      = ws;  ws += 64;
  int BLK1 = (N + CHUNK - 1) / CHUNK;
  float* cval    = ws;  ws += BLK1 * 64;
  int*   cidx    = (int*)ws; ws += BLK1 * 64;
  float* x_tilde = ws;  ws += 64 * 64;
  float* gx      = ws;  ws += 64 * 192;
  float* gh      = ws;  ws += 64 * 192;
  float* Wv      = ws;  ws += 64 * 64;

  hipMemsetAsync(deg,  0, (size_t)N * sizeof(float), stream);
  hipMemsetAsync(tbuf, 0, (size_t)N * sizeof(float), stream);
  hipMemsetAsync(svec, 0, 64 * sizeof(float), stream);

  prep_pw_kernel<<<1, 64, 0, stream>>>(pool_w, pw);
  score_kernel<<<(N + 255) / 256, 256, 0, stream>>>(x, pw, score, N);
  deg_kernel<<<(E + 255) / 256, 256, 0, stream>>>(col, deg, E);
  dis_kernel<<<(N + 255) / 256, 256, 0, stream>>>(deg, dis, N);
  t_kernel<<<(E + 255) / 256, 256, 0, stream>>>(row, col, dis, tbuf, E);
  wsum_kernel<<<512, 256, 0, stream>>>(x, dis, tbuf, svec, N);
  topk_stage1<<<BLK1, 256, 0, stream>>>(score, N, cval, cidx);
  topk_stage2<<<1, 256, 0, stream>>>(x, cval, cidx, BLK1 * 64, x_tilde);
  gru_wmma_kernel<<<1, 256, 0, stream>>>(x_tilde, init_w, w_ih, w_hh, b_ih, b_hh, gx, gh);
  evolve_kernel<<<16, 256, 0, stream>>>(gx, gh, init_w, Wv);
  tail_kernel<<<1, 256, 0, stream>>>(svec, Wv, N, gnn_w, gnn_b, doc, doc_w, doc_b,
                                     ln_g, ln_b, fus_w, fus_b, task_w, task_b,
                                     time_w, time_b, (float*)d_out);
}